// PointNetPP_73718818669278
// MI455X (gfx1250) — compile-verified
//
#include <hip/hip_runtime.h>
#include <cmath>

#define B_ 8
#define TDIM_ 128
#define KC 16   // K-chunk staged through LDS per block iteration

typedef float v2f __attribute__((ext_vector_type(2)));
typedef float v8f __attribute__((ext_vector_type(8)));

static __device__ __forceinline__ v8f wmma4(v2f a, v2f b, v8f c) {
#if __has_builtin(__builtin_amdgcn_wmma_f32_16x16x4_f32)
  return __builtin_amdgcn_wmma_f32_16x16x4_f32(false, a, false, b, (short)0, c, false, false);
#else
  c[0] += a.x * b.x;  // fallback placeholder (should not trigger on gfx1250)
  return c;
#endif
}

// ---------------------------------------------------------------------------
// WMMA f32 GEMM:  C[z][M,N] = alpha * op(A)[M,K] * op(B)[K,N] (+ bias[M])
//  TA=0: A[m,k] = A[m*lda+k]   TA=1: A[m,k] = A[k*lda+m]
//  TB=0: B[k,n] = B[k*ldb+n]   TB=1: B[k,n] = B[n*ldb+k]
// block = 4 waves; block tile = 64(M) x 64(N).
// B tile (KC x 64) staged in LDS (shared by all 4 waves, reused by 4 N-tiles),
// A fragment held in regs and reused across the 4 accumulators.
// ---------------------------------------------------------------------------
template<bool TA, bool TB>
__global__ __launch_bounds__(128)
void k_gemm(const float* __restrict__ A, const float* __restrict__ Bm,
            const float* __restrict__ bias, float* __restrict__ C,
            int M, int N, int K, long lda, long ldb, long ldc,
            long sA, long sB, long sC, float alpha)
{
  __shared__ float Bs[KC][65];                 // +1 pad: kills half-wave bank conflicts
  A  += (long)blockIdx.z * sA;
  Bm += (long)blockIdx.z * sB;
  C  += (long)blockIdx.z * sC;
  const int wave = threadIdx.x >> 5;
  const int lane = threadIdx.x & 31;
  const int half = lane >> 4;
  const int sub  = lane & 15;
  const int tm  = (blockIdx.x * 4 + wave) * 16;
  const int tn0 = blockIdx.y * 64;
  const bool wact = tm < M;                    // no early return: all waves hit barriers
  int am = tm + sub; if (am > M - 1) am = M - 1;
  v8f acc0 = {}, acc1 = {}, acc2 = {}, acc3 = {};

  for (int kb = 0; kb < K; kb += KC) {
    // cooperative, zero-padded B tile load (handles K and N tails uniformly)
    for (int idx = threadIdx.x; idx < KC * 64; idx += 128) {
      const int kk = idx >> 6, nn = idx & 63;
      const int kg = kb + kk, ng = tn0 + nn;
      float v = 0.f;
      if (kg < K && ng < N)
        v = TB ? Bm[(long)ng * ldb + kg] : Bm[(long)kg * ldb + ng];
      Bs[kk][nn] = v;
      const int kg2 = kg + KC;                 // prefetch next chunk into cache
      if (kg2 < K && ng < N)
        __builtin_prefetch(TB ? &Bm[(long)ng * ldb + kg2] : &Bm[(long)kg2 * ldb + ng], 0, 3);
    }
    __syncthreads();
#pragma unroll
    for (int k4 = 0; k4 < KC; k4 += 4) {
      const int ka = kb + k4 + half * 2;       // A fragment (global, row-clamped, K zero-pad)
      v2f a;
      a.x = (ka     < K) ? (TA ? A[(long)ka * lda + am]       : A[(long)am * lda + ka])     : 0.f;
      a.y = (ka + 1 < K) ? (TA ? A[(long)(ka + 1) * lda + am] : A[(long)am * lda + ka + 1]) : 0.f;
      const int kr = k4 + half * 2;            // B fragments from LDS
      v2f b0, b1, b2, b3;
      b0.x = Bs[kr][sub];      b0.y = Bs[kr + 1][sub];
      b1.x = Bs[kr][sub + 16]; b1.y = Bs[kr + 1][sub + 16];
      b2.x = Bs[kr][sub + 32]; b2.y = Bs[kr + 1][sub + 32];
      b3.x = Bs[kr][sub + 48]; b3.y = Bs[kr + 1][sub + 48];
      acc0 = wmma4(a, b0, acc0);
      acc1 = wmma4(a, b1, acc1);
      acc2 = wmma4(a, b2, acc2);
      acc3 = wmma4(a, b3, acc3);
    }
    __syncthreads();
  }

  if (wact) {
#pragma unroll
    for (int t = 0; t < 4; ++t) {
      const v8f acc = t == 0 ? acc0 : t == 1 ? acc1 : t == 2 ? acc2 : acc3;
      const int col = tn0 + 16 * t + sub;
      if (col < N) {
#pragma unroll
        for (int r = 0; r < 8; ++r) {
          const int row = tm + half * 8 + r;   // C layout: vgpr r = M row r (+8 hi half)
          if (row < M) {
            float v = acc[r] * alpha;
            if (bias) v += bias[row];
            C[(long)row * ldc + col] = v;
          }
        }
      }
    }
  }
}

// ---------------------------------------------------------------------------
// elementwise / reduction kernels
// ---------------------------------------------------------------------------
#define GS long i0 = (long)blockIdx.x * blockDim.x + threadIdx.x; \
           long gs = (long)gridDim.x * blockDim.x;

__global__ void k_add(float* y, const float* x, long n) {
  GS for (long i = i0; i < n; i += gs) y[i] += x[i];
}

__global__ void k_copy_ch(const float* __restrict__ s, float* __restrict__ d,
                          int C, int L, long sbs, long dbs) {
  long n = (long)B_ * C * L; GS
  for (long i = i0; i < n; i += gs) {
    long l = i % L; long t = i / L; int c = (int)(t % C); int b = (int)(t / C);
    d[(long)b * dbs + (long)c * L + l] = s[(long)b * sbs + (long)c * L + l];
  }
}

__global__ void k_bn_act(float* x, const float* g, const float* bb, const float* mm,
                         const float* vv, int C, long L, long bstr, float slope, int act) {
  long n = (long)B_ * C * L; GS
  for (long i = i0; i < n; i += gs) {
    long l = i % L; long t = i / L; int c = (int)(t % C); int b = (int)(t / C);
    float* p = x + (long)b * bstr + (long)c * L + l;
    float y = (*p - mm[c]) * (g[c] * rsqrtf(vv[c] + 1e-5f)) + bb[c];
    if (act) y = y >= 0.f ? y : slope * y;
    *p = y;
  }
}

__global__ void k_addtime(float* x, const float* t, int C, long L, long bstr) {
  long n = (long)B_ * C * L; GS
  for (long i = i0; i < n; i += gs) {
    long l = i % L; long q = i / L; int c = (int)(q % C); int b = (int)(q / C);
    x[(long)b * bstr + (long)c * L + l] += t[(long)b * C + c];
  }
}

__global__ void k_dense(const float* __restrict__ x, const float* __restrict__ W,
                        const float* __restrict__ bias, float* __restrict__ y,
                        int O, int K, int act) {
  long n = (long)B_ * O; GS
  for (long i = i0; i < n; i += gs) {
    int o = (int)(i % O); int b = (int)(i / O);
    const float* xp = x + (long)b * K;
    const float* wp = W + (long)o * K;
    float a = bias[o];
    for (int k = 0; k < K; ++k) a += xp[k] * wp[k];
    if (act) a = a / (1.f + expf(-a));   // silu
    y[i] = a;
  }
}

__global__ void k_softmax_row(float* w, long rows, int n) {
  GS
  for (long r = i0; r < rows; r += gs) {
    float* p = w + r * (long)n;
    float mx = -3.4e38f;
    for (int j = 0; j < n; ++j) mx = fmaxf(mx, p[j]);
    float s = 0.f;
    for (int j = 0; j < n; ++j) { float e = expf(p[j] - mx); p[j] = e; s += e; }
    float inv = 1.f / s;
    for (int j = 0; j < n; ++j) p[j] *= inv;
  }
}

// farthest point sampling, one block per batch; dist + centroid vec in LDS
__global__ void k_fps(const float* __restrict__ sup, int C, int Np, int M, int* __restrict__ out) {
  const int b = blockIdx.x;
  sup += (long)b * C * Np;
  extern __shared__ float sm[];
  float* dist = sm;                 // Np
  float* cvec = sm + Np;            // C
  float* red  = cvec + C;           // 256
  int*   redi = (int*)(red + 256);  // 256
  __shared__ int s_far;
  const int t = threadIdx.x;
  for (int n = t; n < Np; n += blockDim.x) dist[n] = 1e10f;
  if (t == 0) s_far = 0;
  __syncthreads();
  for (int it = 0; it < M; ++it) {
    const int far = s_far;
    if (t == 0) out[(long)b * M + it] = far;
    for (int c = t; c < C; c += blockDim.x) cvec[c] = sup[(long)c * Np + far];
    __syncthreads();
    float bestd = -1.f; int besti = 0;
    for (int n = t; n < Np; n += blockDim.x) {
      float d = 0.f;
      for (int c = 0; c < C; ++c) { float df = sup[(long)c * Np + n] - cvec[c]; d += df * df; }
      float dn = fminf(dist[n], d);
      dist[n] = dn;
      if (dn > bestd) { bestd = dn; besti = n; }
    }
    red[t] = bestd; redi[t] = besti;
    __syncthreads();
    for (int s = 128; s > 0; s >>= 1) {
      if (t < s) {
        if (red[t + s] > red[t] || (red[t + s] == red[t] && redi[t + s] < redi[t])) {
          red[t] = red[t + s]; redi[t] = redi[t + s];
        }
      }
      __syncthreads();
    }
    if (t == 0) s_far = redi[0];
    __syncthreads();
  }
}

// ball query: one block per (m, b); flags in LDS, ordered first-k scan
__global__ void k_ballq(const float* __restrict__ sup, const float* __restrict__ cent,
                        int C, int Np, int M, int Kn, float r2,
                        int* __restrict__ idx, int* __restrict__ msk) {
  const int m = blockIdx.x, b = blockIdx.y;
  sup  += (long)b * C * Np;
  cent += (long)b * C * M;
  extern __shared__ float sm[];
  float* cvec = sm;               // C
  int*   flg  = (int*)(sm + C);   // Np
  const int t = threadIdx.x;
  for (int c = t; c < C; c += blockDim.x) cvec[c] = cent[(long)c * M + m];
  __syncthreads();
  for (int n = t; n < Np; n += blockDim.x) {
    float d = 0.f;
    for (int c = 0; c < C; ++c) { float df = sup[(long)c * Np + n] - cvec[c]; d += df * df; }
    flg[n] = d <= r2;
  }
  __syncthreads();
  if (t == 0) {
    const long base = ((long)b * M + m) * Kn;
    int cnt = 0;
    for (int n = 0; n < Np && cnt < Kn; ++n)
      if (flg[n]) { idx[base + cnt] = n; msk[base + cnt] = 0; ++cnt; }
    const int f0 = cnt ? idx[base] : 0;
    for (int j = cnt; j < Kn; ++j) { idx[base + j] = f0; msk[base + j] = 1; }
  }
}

__global__ void k_gather(const float* __restrict__ sup, const int* __restrict__ cidx,
                         float* __restrict__ dst, int C, int Np, int M) {
  long n = (long)B_ * C * M; GS
  for (long i = i0; i < n; i += gs) {
    int m = (int)(i % M); long t = i / M; int c = (int)(t % C); int b = (int)(t / C);
    dst[i] = sup[((long)b * C + c) * Np + cidx[(long)b * M + m]];
  }
}

__global__ void k_group(const float* __restrict__ sup, const float* __restrict__ cent,
                        const int* __restrict__ gidx, float* __restrict__ grp,
                        int C, int Np, int M, int Kn) {
  long n = (long)B_ * C * M * Kn; GS
  for (long i = i0; i < n; i += gs) {
    int j = (int)(i % Kn); long t = i / Kn; int m = (int)(t % M);
    t /= M; int c = (int)(t % C); int b = (int)(t / C);
    grp[i] = sup[((long)b * C + c) * Np + gidx[((long)b * M + m) * Kn + j]]
           - cent[((long)b * C + c) * M + m];
  }
}

// attn-mix logits: w[b,h,m,j] = scale * sum_c q[b,hC+c,m] * k[b,hC+c,m,j] ; mask -> -1e9
__global__ void k_amw(const float* __restrict__ q, const float* __restrict__ kk,
                      const int* __restrict__ msk, float* __restrict__ w,
                      int C, int M, int Kn, float scale) {
  long n = (long)B_ * 2 * M * Kn; GS
  for (long i = i0; i < n; i += gs) {
    int j = (int)(i % Kn); long t = i / Kn; int m = (int)(t % M);
    t /= M; int h = (int)(t % 2); int b = (int)(t / 2);
    const float* qp = q  + ((long)b * 2 * C + (long)h * C) * M + m;
    const float* kp = kk + ((long)b * 2 * C + (long)h * C) * (long)M * Kn + (long)m * Kn + j;
    float a = 0.f;
    for (int c = 0; c < C; ++c) a += qp[(long)c * M] * kp[(long)c * M * Kn];
    w[i] = msk[((long)b * M + m) * Kn + j] ? -1e9f : a * scale;
  }
}

// attn-mix output: o[b,ch,m] = sum_j w[b,ch/C,m,j] * v[b,ch,m,j]
__global__ void k_amo(const float* __restrict__ w, const float* __restrict__ vv,
                      float* __restrict__ o, int C, int M, int Kn) {
  long n = (long)B_ * 2 * C * M; GS
  for (long i = i0; i < n; i += gs) {
    int m = (int)(i % M); long t = i / M; int ch = (int)(t % (2 * C)); int b = (int)(t / (2 * C));
    const int h = ch / C;
    const float* wp = w  + (((long)b * 2 + h) * M + m) * Kn;
    const float* vp = vv + ((long)b * 2 * C + ch) * (long)M * Kn + (long)m * Kn;
    float a = 0.f;
    for (int j = 0; j < Kn; ++j) a += wp[j] * vp[j];
    o[i] = a;
  }
}

// ---------------------------------------------------------------------------
// host side
// ---------------------------------------------------------------------------
struct Cur { void* const* p; int i; const float* f() { return (const float*)p[i++]; } };
struct ConvP { const float *W, *b, *g, *bb, *m, *v; int Co; };
struct TmlpP { ConvP cv[3]; int n; const float *tW1, *tb1, *tW2, *tb2; };
struct AttnP { const float *q, *k, *v, *o; };
struct ScaleP { AttnP at; TmlpP mlp; };
struct UpP { AttnP at; TmlpP mlp; };

struct Ar {
  char* base; size_t off;
  float* f(size_t n) { float* p = (float*)(base + off); off += ((n * 4 + 255) & ~(size_t)255); return p; }
  int*   i(size_t n) { int*   p = (int*)(base + off);   off += ((n * 4 + 255) & ~(size_t)255); return p; }
};

static inline dim3 eg(long n) {
  long b = (n + 255) / 256; if (b < 1) b = 1; if (b > 262144) b = 262144;
  return dim3((unsigned)b);
}

static void gemm(hipStream_t st, int ta, int tb, const float* A, const float* Bm,
                 const float* bias, float* C, int M, int N, int K,
                 long lda, long ldb, long ldc, long sA, long sB, long sC,
                 int nz, float alpha = 1.f) {
  dim3 g((M + 63) / 64, (N + 63) / 64, nz), b(128);
  if (ta) {
    if (tb) k_gemm<true, true ><<<g, b, 0, st>>>(A, Bm, bias, C, M, N, K, lda, ldb, ldc, sA, sB, sC, alpha);
    else    k_gemm<true, false><<<g, b, 0, st>>>(A, Bm, bias, C, M, N, K, lda, ldb, ldc, sA, sB, sC, alpha);
  } else {
    if (tb) k_gemm<false, true ><<<g, b, 0, st>>>(A, Bm, bias, C, M, N, K, lda, ldb, ldc, sA, sB, sC, alpha);
    else    k_gemm<false, false><<<g, b, 0, st>>>(A, Bm, bias, C, M, N, K, lda, ldb, ldc, sA, sB, sC, alpha);
  }
}

static AttnP readAttn(Cur& c) { AttnP a; a.q = c.f(); a.k = c.f(); a.v = c.f(); a.o = c.f(); return a; }
static TmlpP readTmlp(Cur& c, const int* hids, int n) {
  TmlpP t; t.n = n;
  for (int j = 0; j < n; ++j) {
    t.cv[j].W = c.f(); t.cv[j].b = c.f();
    t.cv[j].g = c.f(); t.cv[j].bb = c.f(); t.cv[j].m = c.f(); t.cv[j].v = c.f();
    t.cv[j].Co = hids[j];
  }
  t.tW1 = c.f(); t.tb1 = c.f(); t.tW2 = c.f(); t.tb2 = c.f();
  return t;
}

// TemporalMLP: conv+bn (+leaky 0.02 except last) then add time embedding
static void run_tmlp(hipStream_t st, Ar& ar, const TmlpP& p, const float* x, int Cin,
                     long xbstr, int L, float* outBase, long outBstr, const float* temb) {
  const float* cur = x; long curb = xbstr; int Ci = Cin;
  for (int j = 0; j < p.n; ++j) {
    const int Co = p.cv[j].Co;
    float* dst; long db;
    if (j == p.n - 1) { dst = outBase; db = outBstr; }
    else { dst = ar.f((size_t)B_ * Co * L); db = (long)Co * L; }
    gemm(st, 0, 0, p.cv[j].W, cur, p.cv[j].b, dst, Co, L, Ci,
         Ci, L, L, 0, curb, db, B_);
    k_bn_act<<<eg((long)B_ * Co * L), 256, 0, st>>>(dst, p.cv[j].g, p.cv[j].bb, p.cv[j].m,
                                                    p.cv[j].v, Co, L, db, 0.02f, j < p.n - 1);
    cur = dst; curb = db; Ci = Co;
  }
  float* t1 = ar.f((size_t)B_ * 512);
  k_dense<<<eg((long)B_ * 512), 256, 0, st>>>(temb, p.tW1, p.tb1, t1, 512, TDIM_, 1);
  float* t2 = ar.f((size_t)B_ * Ci);
  k_dense<<<eg((long)B_ * Ci), 256, 0, st>>>(t1, p.tW2, p.tb2, t2, Ci, 512, 0);
  k_addtime<<<eg((long)B_ * Ci * L), 256, 0, st>>>(outBase, t2, Ci, L, outBstr);
}

// one MSG set-abstraction level; writes xyz (ch 0..2) + features (ch 3..) into dsup
static void run_msg(hipStream_t st, Ar& ar, const float* sup, int C, int Np, int M,
                    const float* rad, const int* ks, const ScaleP* sp,
                    const float* temb, float* dsup, int Ctot) {
  const size_t save = ar.off;
  int* cidx = ar.i((size_t)B_ * M);
  {
    size_t sh = ((size_t)Np + C + 512) * sizeof(float);
    k_fps<<<dim3(B_), dim3(256), sh, st>>>(sup, C, Np, M, cidx);
  }
  float* cent = ar.f((size_t)B_ * C * M);
  k_gather<<<eg((long)B_ * C * M), 256, 0, st>>>(sup, cidx, cent, C, Np, M);
  k_copy_ch<<<eg((long)B_ * 3 * M), 256, 0, st>>>(cent, dsup, 3, M, (long)C * M, (long)Ctot * M);
  int co = 3;
  const int C2 = 2 * C;
  for (int s = 0; s < 2; ++s) {
    const size_t save2 = ar.off;
    const int Kn = ks[s];
    const float r2 = rad[s] * rad[s];
    int* idx = ar.i((size_t)B_ * M * Kn);
    int* msk = ar.i((size_t)B_ * M * Kn);
    {
      size_t sh = ((size_t)C + Np) * sizeof(float);
      k_ballq<<<dim3(M, B_), 256, sh, st>>>(sup, cent, C, Np, M, Kn, r2, idx, msk);
    }
    float* grp = ar.f((size_t)B_ * C * M * Kn);
    k_group<<<eg((long)B_ * C * M * Kn), 256, 0, st>>>(sup, cent, idx, grp, C, Np, M, Kn);
    const long LG = (long)M * Kn;
    float* q = ar.f((size_t)B_ * C2 * M);
    gemm(st, 0, 0, sp[s].at.q, cent, nullptr, q, C2, M, C, C, M, M, 0, (long)C * M, (long)C2 * M, B_);
    float* kk = ar.f((size_t)B_ * C2 * LG);
    gemm(st, 0, 0, sp[s].at.k, grp, nullptr, kk, C2, (int)LG, C, C, LG, LG, 0, (long)C * LG, (long)C2 * LG, B_);
    float* vv = ar.f((size_t)B_ * C2 * LG);
    gemm(st, 0, 0, sp[s].at.v, grp, nullptr, vv, C2, (int)LG, C, C, LG, LG, 0, (long)C * LG, (long)C2 * LG, B_);
    float* w = ar.f((size_t)B_ * 2 * M * Kn);
    k_amw<<<eg((long)B_ * 2 * M * Kn), 256, 0, st>>>(q, kk, msk, w, C, M, Kn, 1.f / sqrtf((float)C));
    k_softmax_row<<<eg((long)B_ * 2 * M), 256, 0, st>>>(w, (long)B_ * 2 * M, Kn);
    float* o = ar.f((size_t)B_ * C2 * M);
    k_amo<<<eg((long)B_ * C2 * M), 256, 0, st>>>(w, vv, o, C, M, Kn);
    float* fa = ar.f((size_t)B_ * C * M);
    gemm(st, 0, 0, sp[s].at.o, o, nullptr, fa, C, M, C2, C2, M, M, 0, (long)C2 * M, (long)C * M, B_);
    k_add<<<eg((long)B_ * C * M), 256, 0, st>>>(fa, cent, (long)B_ * C * M);
    run_tmlp(st, ar, sp[s].mlp, fa, C, (long)C * M, M, dsup + (long)co * M, (long)Ctot * M, temb);
    co += sp[s].mlp.cv[sp[s].mlp.n - 1].Co;
    ar.off = save2;
  }
  ar.off = save;
}

// feature propagation: Attention1d (WMMA batched q^T k and p v^T) + concat + tmlp
static void run_fp(hipStream_t st, Ar& ar, const UpP& p, const float* xin, int Cq, int Nx,
                   const float* yin, int Ck, int Ny, int oc,
                   const float* xf, int Cxf, long xfbstr,
                   float* fout, int Cout, const float* temb) {
  const size_t save = ar.off;
  const int O2 = 2 * oc;
  float* q = ar.f((size_t)B_ * O2 * Nx);
  gemm(st, 0, 0, p.at.q, xin, nullptr, q, O2, Nx, Cq, Cq, Nx, Nx, 0, (long)Cq * Nx, (long)O2 * Nx, B_);
  float* kk = ar.f((size_t)B_ * O2 * Ny);
  gemm(st, 0, 0, p.at.k, yin, nullptr, kk, O2, Ny, Ck, Ck, Ny, Ny, 0, (long)Ck * Ny, (long)O2 * Ny, B_);
  float* vv = ar.f((size_t)B_ * O2 * Ny);
  gemm(st, 0, 0, p.at.v, yin, nullptr, vv, O2, Ny, Ck, Ck, Ny, Ny, 0, (long)Ck * Ny, (long)O2 * Ny, B_);
  float* lg = ar.f((size_t)B_ * 2 * Nx * Ny);
  gemm(st, 1, 0, q, kk, nullptr, lg, Nx, Ny, oc,
       Nx, Ny, Ny, (long)oc * Nx, (long)oc * Ny, (long)Nx * Ny, 2 * B_, 1.f / sqrtf((float)oc));
  k_softmax_row<<<eg((long)2 * B_ * Nx), 256, 0, st>>>(lg, (long)2 * B_ * Nx, Ny);
  float* ao = ar.f((size_t)B_ * O2 * Nx);
  gemm(st, 0, 1, vv, lg, nullptr, ao, oc, Nx, Ny,
       Ny, Ny, Nx, (long)oc * Ny, (long)Nx * Ny, (long)oc * Nx, 2 * B_);
  const int Cmid = oc + (xf ? Cxf : 0);
  float* mi = ar.f((size_t)B_ * Cmid * Nx);
  gemm(st, 0, 0, p.at.o, ao, nullptr, mi, oc, Nx, O2, O2, Nx, Nx, 0, (long)O2 * Nx, (long)Cmid * Nx, B_);
  if (xf)
    k_copy_ch<<<eg((long)B_ * Cxf * Nx), 256, 0, st>>>(xf, mi + (long)oc * Nx, Cxf, Nx,
                                                       xfbstr, (long)Cmid * Nx);
  run_tmlp(st, ar, p.mlp, mi, Cmid, (long)Cmid * Nx, Nx, fout, (long)Cout * Nx, temb);
  ar.off = save;
}

extern "C" void kernel_launch(void* const* d_in, const int* in_sizes, int n_in,
                              void* d_out, int out_size, void* d_ws, size_t ws_size,
                              hipStream_t stream) {
  const float* x    = (const float*)d_in[0];   // [8,3,2048]
  const float* temb = (const float*)d_in[1];   // [8,128]
  Cur cur{d_in, 2};

  static const int dh[4][2][3] = {
    {{16, 16, 32}, {32, 32, 64}},
    {{64, 64, 128}, {64, 96, 128}},
    {{128, 196, 256}, {128, 196, 256}},
    {{256, 256, 512}, {256, 384, 512}}};
  ScaleP down[4][2];
  for (int l = 0; l < 4; ++l)
    for (int s = 0; s < 2; ++s) {
      down[l][s].at = readAttn(cur);
      down[l][s].mlp = readTmlp(cur, dh[l][s], 3);
    }
  static const int uh[4][3] = {{256, 256, 0}, {256, 256, 0}, {256, 128, 0}, {128, 128, 128}};
  static const int un[4] = {2, 2, 2, 3};
  UpP up[4];
  for (int l = 0; l < 4; ++l) {
    up[l].at = readAttn(cur);
    up[l].mlp = readTmlp(cur, uh[l], un[l]);
  }
  const float* pW1 = cur.f(); const float* pb1 = cur.f();
  const float* pg = cur.f(); const float* pbb = cur.f();
  const float* pm = cur.f(); const float* pv = cur.f();
  const float* pW2 = cur.f(); const float* pb2 = cur.f();

  Ar ar{(char*)d_ws, 0};
  float* sup2 = ar.f((size_t)B_ * 99 * 1024);
  float* sup3 = ar.f((size_t)B_ * 259 * 512);
  float* sup4 = ar.f((size_t)B_ * 515 * 128);
  float* sup5 = ar.f((size_t)B_ * 1027 * 64);
  float* f3u  = ar.f((size_t)B_ * 256 * 128);
  float* f2u  = ar.f((size_t)B_ * 256 * 512);
  float* f1u  = ar.f((size_t)B_ * 128 * 1024);
  float* f0   = ar.f((size_t)B_ * 128 * 2048);

  static const float rr[4][2] = {{0.05f, 0.1f}, {0.1f, 0.2f}, {0.2f, 0.4f}, {0.4f, 0.8f}};
  static const int kk2[2] = {16, 32};

  run_msg(stream, ar, x,    3,   2048, 1024, rr[0], kk2, down[0], temb, sup2, 99);
  run_msg(stream, ar, sup2, 99,  1024, 512,  rr[1], kk2, down[1], temb, sup3, 259);
  run_msg(stream, ar, sup3, 259, 512,  128,  rr[2], kk2, down[2], temb, sup4, 515);
  run_msg(stream, ar, sup4, 515, 128,  64,   rr[3], kk2, down[3], temb, sup5, 1027);

  // up1: xin=sup4 [515,128], yin=sup5 [1027,64], xf = f3 view (sup4 ch3..)
  run_fp(stream, ar, up[0], sup4, 515, 128, sup5, 1027, 64, 1024,
         sup4 + (long)3 * 128, 512, (long)515 * 128, f3u, 256, temb);

  { // up2: yin = concat(xyz3, f3u) [259,128]
    size_t sv = ar.off;
    float* yb = ar.f((size_t)B_ * 259 * 128);
    k_copy_ch<<<eg((long)B_ * 3 * 128), 256, 0, stream>>>(sup4, yb, 3, 128, (long)515 * 128, (long)259 * 128);
    k_copy_ch<<<eg((long)B_ * 256 * 128), 256, 0, stream>>>(f3u, yb + (long)3 * 128, 256, 128,
                                                            (long)256 * 128, (long)259 * 128);
    run_fp(stream, ar, up[1], sup3, 259, 512, yb, 259, 128, 256,
           sup3 + (long)3 * 512, 256, (long)259 * 512, f2u, 256, temb);
    ar.off = sv;
  }
  { // up3: yin = concat(xyz2, f2u) [259,512]
    size_t sv = ar.off;
    float* yb = ar.f((size_t)B_ * 259 * 512);
    k_copy_ch<<<eg((long)B_ * 3 * 512), 256, 0, stream>>>(sup3, yb, 3, 512, (long)259 * 512, (long)259 * 512);
    k_copy_ch<<<eg((long)B_ * 256 * 512), 256, 0, stream>>>(f2u, yb + (long)3 * 512, 256, 512,
                                                            (long)256 * 512, (long)259 * 512);
    run_fp(stream, ar, up[2], sup2, 99, 1024, yb, 259, 512, 256,
           sup2 + (long)3 * 1024, 96, (long)99 * 1024, f1u, 128, temb);
    ar.off = sv;
  }
  { // up4: yin = concat(xyz1, f1u) [131,1024]
    size_t sv = ar.off;
    float* yb = ar.f((size_t)B_ * 131 * 1024);
    k_copy_ch<<<eg((long)B_ * 3 * 1024), 256, 0, stream>>>(sup2, yb, 3, 1024, (long)99 * 1024, (long)131 * 1024);
    k_copy_ch<<<eg((long)B_ * 128 * 1024), 256, 0, stream>>>(f1u, yb + (long)3 * 1024, 128, 1024,
                                                             (long)128 * 1024, (long)131 * 1024);
    run_fp(stream, ar, up[3], x, 3, 2048, yb, 131, 1024, 128,
           nullptr, 0, 0, f0, 128, temb);
    ar.off = sv;
  }
  { // prediction head: conv(128->1024)+bn+leaky(0.01) then conv(1024->3)
    size_t sv = ar.off;
    float* h = ar.f((size_t)B_ * 1024 * 2048);
    gemm(stream, 0, 0, pW1, f0, pb1, h, 1024, 2048, 128,
         128, 2048, 2048, 0, (long)128 * 2048, (long)1024 * 2048, B_);
    k_bn_act<<<eg((long)B_ * 1024 * 2048), 256, 0, stream>>>(h, pg, pbb, pm, pv, 1024, 2048,
                                                             (long)1024 * 2048, 0.01f, 1);
    gemm(stream, 0, 0, pW2, h, pb2, (float*)d_out, 3, 2048, 1024,
         1024, 2048, 2048, 0, (long)1024 * 2048, (long)3 * 2048, B_);
    ar.off = sv;
  }
}